// FAIIAHead_10788957848045
// MI455X (gfx1250) — compile-verified
//
#include <hip/hip_runtime.h>
#include <hip/hip_bf16.h>

typedef __attribute__((ext_vector_type(2))) float v2f;
typedef __attribute__((ext_vector_type(8))) float v8f;

#define BROWS   262144
#define DIN     512
#define DATT    32
#define NPROTO  4
#define KSTEPS  (DIN / 4)        // 128 WMMA k-steps of K=4
#define ROWS_PER_WAVE 64         // 4 M-tiles of 16
#define WAVES_PER_WG  8
#define ROWS_PER_WG   (ROWS_PER_WAVE * WAVES_PER_WG)   // 512
#define LDS_PAD 33               // floats per staged row (conflict-free)
#define SCALE_F 0.17677669529663687f  // 32^-0.5
#define LN_EPS  1e-5f

__global__ __launch_bounds__(256)
void faiia_head_kernel(const float* __restrict__ x,
                       const float* __restrict__ minority_prob,
                       const float* __restrict__ Wq,
                       const float* __restrict__ bq,
                       const float* __restrict__ proto_keys,
                       const float* __restrict__ proto_values,
                       const float* __restrict__ proto_importance,
                       const float* __restrict__ focal_alpha,
                       const float* __restrict__ focal_temp,
                       const float* __restrict__ Wo,
                       const float* __restrict__ bo,
                       const float* __restrict__ ln_gamma,
                       const float* __restrict__ ln_beta,
                       float* __restrict__ out,
                       float* __restrict__ weights_out)
{
    // Per-wave private staging: 32 rows x 33 floats, 8 waves -> 33792 B
    __shared__ float qstage[WAVES_PER_WG * 32 * LDS_PAD];

    const int tid  = threadIdx.x;
    const int wave = tid >> 5;
    const int lane = tid & 31;
    const int half = lane >> 4;        // 0: lanes 0-15, 1: lanes 16-31
    const int l16  = lane & 15;

    const long rowBase = (long)blockIdx.x * ROWS_PER_WG + (long)wave * ROWS_PER_WAVE;

    // ---- main GEMM: q = x @ Wq^T  via V_WMMA_F32_16X16X4_F32 ----
    v8f acc[4][2] = {};   // [mtile][ntile], each 16x16 f32 tile

    // A fragment: lane holds x[row = base + mt*16 + l16][4k + 2*half + {0,1}]
    const float* aptr0 = x + (rowBase +  0 + l16) * (long)DIN + 2 * half;
    const float* aptr1 = x + (rowBase + 16 + l16) * (long)DIN + 2 * half;
    const float* aptr2 = x + (rowBase + 32 + l16) * (long)DIN + 2 * half;
    const float* aptr3 = x + (rowBase + 48 + l16) * (long)DIN + 2 * half;
    // B fragment: lane holds Wq[n = nt*16 + l16][4k + 2*half + {0,1}]
    const float* bptr0 = Wq + ( 0 + l16) * DIN + 2 * half;
    const float* bptr1 = Wq + (16 + l16) * DIN + 2 * half;

    #pragma unroll 4
    for (int k = 0; k < KSTEPS; ++k) {
        const int off = k * 4;
        v2f bf0 = *(const v2f*)(bptr0 + off);
        v2f bf1 = *(const v2f*)(bptr1 + off);
        v2f a0  = *(const v2f*)(aptr0 + off);
        v2f a1  = *(const v2f*)(aptr1 + off);
        v2f a2  = *(const v2f*)(aptr2 + off);
        v2f a3  = *(const v2f*)(aptr3 + off);

        acc[0][0] = __builtin_amdgcn_wmma_f32_16x16x4_f32(false, a0, false, bf0, (short)0, acc[0][0], false, false);
        acc[0][1] = __builtin_amdgcn_wmma_f32_16x16x4_f32(false, a0, false, bf1, (short)0, acc[0][1], false, false);
        acc[1][0] = __builtin_amdgcn_wmma_f32_16x16x4_f32(false, a1, false, bf0, (short)0, acc[1][0], false, false);
        acc[1][1] = __builtin_amdgcn_wmma_f32_16x16x4_f32(false, a1, false, bf1, (short)0, acc[1][1], false, false);
        acc[2][0] = __builtin_amdgcn_wmma_f32_16x16x4_f32(false, a2, false, bf0, (short)0, acc[2][0], false, false);
        acc[2][1] = __builtin_amdgcn_wmma_f32_16x16x4_f32(false, a2, false, bf1, (short)0, acc[2][1], false, false);
        acc[3][0] = __builtin_amdgcn_wmma_f32_16x16x4_f32(false, a3, false, bf0, (short)0, acc[3][0], false, false);
        acc[3][1] = __builtin_amdgcn_wmma_f32_16x16x4_f32(false, a3, false, bf1, (short)0, acc[3][1], false, false);
    }

    // scalar broadcast params
    const float alpha = focal_alpha[0];
    const float ftemp = focal_temp[0];

    float* wls = qstage + wave * (32 * LDS_PAD);

    // ---- epilogue in two 32-row chunks (per-wave private LDS, no barriers) ----
    #pragma unroll
    for (int chunk = 0; chunk < 2; ++chunk) {
        // stage 32 rows of q into LDS (C/D layout: VGPR j -> M = j + 8*half, N = l16 + 16*nt)
        #pragma unroll
        for (int mt2 = 0; mt2 < 2; ++mt2) {
            const int mt = chunk * 2 + mt2;
            #pragma unroll
            for (int nt = 0; nt < 2; ++nt) {
                const int col = nt * 16 + l16;
                const float bqv = bq[col];
                #pragma unroll
                for (int j = 0; j < 8; ++j) {
                    const int row = mt2 * 16 + j + 8 * half;
                    wls[row * LDS_PAD + col] = acc[mt][nt][j] + bqv;
                }
            }
        }

        // each lane now owns one full row
        const long grow = rowBase + chunk * 32 + lane;
        const float* qp = wls + lane * LDS_PAD;

        // scores[p] = q . proto_keys[p] + proto_importance[p]
        float s0 = proto_importance[0];
        float s1 = proto_importance[1];
        float s2 = proto_importance[2];
        float s3 = proto_importance[3];
        #pragma unroll 8
        for (int n = 0; n < DATT; ++n) {
            const float qv = qp[n];
            s0 += qv * proto_keys[0 * DATT + n];
            s1 += qv * proto_keys[1 * DATT + n];
            s2 += qv * proto_keys[2 * DATT + n];
            s3 += qv * proto_keys[3 * DATT + n];
        }

        // focal modulation, GAMMA = 2
        const float mp = minority_prob[grow];
        const float u2 = 1.0f - fabsf(mp - 0.5f) * 2.0f + 1e-8f;
        const float fw = alpha * u2 * u2 * ftemp;
        const float mult = (1.0f + fw) * SCALE_F;
        s0 *= mult; s1 *= mult; s2 *= mult; s3 *= mult;

        // softmax over 4 prototypes
        float mx = fmaxf(fmaxf(s0, s1), fmaxf(s2, s3));
        float e0 = __expf(s0 - mx);
        float e1 = __expf(s1 - mx);
        float e2 = __expf(s2 - mx);
        float e3 = __expf(s3 - mx);
        const float rsum = 1.0f / (e0 + e1 + e2 + e3);
        const float w0 = e0 * rsum, w1 = e1 * rsum, w2 = e2 * rsum, w3 = e3 * rsum;

        // attended = weights @ proto_values
        float att[DATT];
        #pragma unroll 8
        for (int n = 0; n < DATT; ++n) {
            att[n] = w0 * proto_values[0 * DATT + n]
                   + w1 * proto_values[1 * DATT + n]
                   + w2 * proto_values[2 * DATT + n]
                   + w3 * proto_values[3 * DATT + n];
        }

        // o = attended @ Wo^T + bo  (Wo accesses are wave-uniform -> scalar loads)
        float o[DATT];
        float mu = 0.0f;
        #pragma unroll 4
        for (int n = 0; n < DATT; ++n) {
            float t = bo[n];
            #pragma unroll 8
            for (int m = 0; m < DATT; ++m)
                t += att[m] * Wo[n * DATT + m];
            o[n] = t;
            mu += t;
        }
        mu *= (1.0f / DATT);
        float var = 0.0f;
        #pragma unroll 8
        for (int n = 0; n < DATT; ++n) {
            const float d = o[n] - mu;
            var += d * d;
        }
        var *= (1.0f / DATT);
        const float inv = rsqrtf(var + LN_EPS);

        float* op = out + (size_t)grow * DATT;
        #pragma unroll 8
        for (int n = 0; n < DATT; ++n)
            op[n] = (o[n] - mu) * inv * ln_gamma[n] + ln_beta[n];

        float* wp = weights_out + (size_t)grow * NPROTO;
        wp[0] = w0; wp[1] = w1; wp[2] = w2; wp[3] = w3;
    }
}

extern "C" void kernel_launch(void* const* d_in, const int* in_sizes, int n_in,
                              void* d_out, int out_size, void* d_ws, size_t ws_size,
                              hipStream_t stream) {
    (void)in_sizes; (void)n_in; (void)d_ws; (void)ws_size; (void)out_size;

    const float* x        = (const float*)d_in[0];
    const float* mprob    = (const float*)d_in[1];
    const float* Wq       = (const float*)d_in[2];
    const float* bq       = (const float*)d_in[3];
    const float* pkeys    = (const float*)d_in[4];
    const float* pvals    = (const float*)d_in[5];
    const float* pimp     = (const float*)d_in[6];
    const float* falpha   = (const float*)d_in[7];
    const float* ftemp    = (const float*)d_in[8];
    const float* Wo       = (const float*)d_in[9];
    const float* bo       = (const float*)d_in[10];
    const float* ln_gamma = (const float*)d_in[11];
    const float* ln_beta  = (const float*)d_in[12];

    float* out  = (float*)d_out;
    float* wout = out + (size_t)BROWS * DATT;   // weights follow `out` in the flat output

    dim3 grid(BROWS / ROWS_PER_WG);   // 512
    dim3 block(256);
    faiia_head_kernel<<<grid, block, 0, stream>>>(
        x, mprob, Wq, bq, pkeys, pvals, pimp, falpha, ftemp,
        Wo, bo, ln_gamma, ln_beta, out, wout);
}